// Hierarchical_Refinement_15238543966317
// MI455X (gfx1250) — compile-verified
//
#include <hip/hip_runtime.h>

typedef float v2f __attribute__((ext_vector_type(2)));
typedef float v8f __attribute__((ext_vector_type(8)));

#define NBATCH   4
#define NCURVE   8192
#define NSURF    24576
#define BPTS     32768            // points per batch
#define GRID_R   128
#define GRID_CELLS (GRID_R*GRID_R*GRID_R)   // 2,097,152
#define CFEAT    16
#define GLO_ELEMS (2*18*18*18)    // 11664 per patch
#define LOC_CELLS (36*36*36)      // 46656 per patch
#define NPATCH   64               // B*P

// ---------------------------------------------------------------- init grid
__global__ __launch_bounds__(256)
void init_grid_k(int4* __restrict__ grid4, int n4) {
    int t = blockIdx.x * blockDim.x + threadIdx.x;
    if (t < n4) grid4[t] = make_int4(-1, -1, -1, -1);
}

// ------------------------------------------------- MLP (WMMA) + cell scatter
// 4 waves per block; each wave owns a 16-point tile.
// Layer1 (3->128, ReLU) in VALU into padded LDS; layer2 (128->16) as a chain
// of 32 V_WMMA_F32_16X16X4_F32.
__global__ __launch_bounds__(128)
void mlp_scatter_k(const float* __restrict__ curves,
                   const float* __restrict__ surfaces,
                   const float* __restrict__ W1, const float* __restrict__ b1,
                   const float* __restrict__ W2, const float* __restrict__ b2,
                   float* __restrict__ feat, int* __restrict__ grid)
{
    __shared__ float pts[4][16][4];       // x,y,z (+pad)
    __shared__ float hl[4][16][132];      // h, pitch 132 -> bank stride 4
    __shared__ float w2s[16][132];        // W2 staged, padded

    const int tid  = threadIdx.x;
    const int lane = tid & 31;
    const int wave = tid >> 5;

    // stage W2 (16x128)
    for (int i = tid; i < 16 * 128; i += 128) {
        int n = i >> 7, k = i & 127;
        w2s[n][k] = W2[n * 128 + k];
    }

    const int tile = blockIdx.x * 4 + wave;   // 16-point tile id
    const int g0   = tile * 16;               // flat point base = b*BPTS + n
    const int b    = g0 / BPTS;               // tiles never straddle batches
    const int n0   = g0 - b * BPTS;

    // ---- Phase A: lanes 0..15 load one point, scatter cell index ----------
    if (lane < 16) {
        int n = n0 + lane;
        const float* src = (n < NCURVE)
            ? (curves   + ((size_t)b * NCURVE + n) * 3)
            : (surfaces + ((size_t)b * NSURF + (n - NCURVE)) * 3);
        float px = src[0], py = src[1], pz = src[2];
        pts[wave][lane][0] = px;
        pts[wave][lane][1] = py;
        pts[wave][lane][2] = pz;
        float cx = fminf(fmaxf(px * 128.f + 64.5f, 0.f), 127.f);
        float cy = fminf(fmaxf(py * 128.f + 64.5f, 0.f), 127.f);
        float cz = fminf(fmaxf(pz * 128.f + 64.5f, 0.f), 127.f);
        int ix = (int)cx, iy = (int)cy, iz = (int)cz;
        // last-write-wins == largest point index wins
        atomicMax(&grid[(size_t)b * GRID_CELLS + ((ix * GRID_R + iy) * GRID_R + iz)], n);
    }
    __syncthreads();

    // ---- Phase B: layer1 (3->128, ReLU) -----------------------------------
    #pragma unroll
    for (int cb = 0; cb < 4; ++cb) {
        int c = cb * 32 + lane;
        float wa = W1[c * 3 + 0], wb = W1[c * 3 + 1], wc = W1[c * 3 + 2];
        float bi = b1[c];
        #pragma unroll
        for (int m = 0; m < 16; ++m) {
            float v = fmaf(pts[wave][m][0], wa,
                      fmaf(pts[wave][m][1], wb,
                      fmaf(pts[wave][m][2], wc, bi)));
            hl[wave][m][c] = fmaxf(v, 0.f);
        }
    }
    __syncthreads();

    // ---- Phase C: layer2 via chained V_WMMA_F32_16X16X4_F32 ---------------
    // A 16x4 f32 layout: lanes 0-15 rows m, K={0,1}; lanes 16-31 rows m, K={2,3}
    const int mn   = lane & 15;           // row for A / column for B / N for D
    const int koff = (lane >> 4) << 1;    // 0 or 2
    v8f acc = {};
    #pragma unroll
    for (int t = 0; t < 32; ++t) {
        int k0 = t * 4 + koff;
        v2f a, bf;
        a.x  = hl[wave][mn][k0];
        a.y  = hl[wave][mn][k0 + 1];
        bf.x = w2s[mn][k0];               // B = W2^T : B[k][n] = W2[n][k]
        bf.y = w2s[mn][k0 + 1];
        acc = __builtin_amdgcn_wmma_f32_16x16x4_f32(
            false, a, false, bf, (short)0, acc, false, false);
    }
    // D layout: lane%16 = N, VGPR r -> M = r + 8*(lane>=16)
    float bias = b2[mn];
    const int mbase = (lane >> 4) * 8;
    #pragma unroll
    for (int r = 0; r < 8; ++r)
        feat[(size_t)(g0 + mbase + r) * CFEAT + mn] = acc[r] + bias;
}

// ------------------------------------------------------------ global patches
__global__ __launch_bounds__(256)
void global_patches_k(const float* __restrict__ occ, const int* __restrict__ idxarr,
                      float* __restrict__ out)
{
    int t = blockIdx.x * blockDim.x + threadIdx.x;
    if (t >= NPATCH * GLO_ELEMS) return;
    int patch = t / GLO_ELEMS;
    int rem   = t - patch * GLO_ELEMS;
    int ch = rem / (18 * 18 * 18);
    int r2 = rem % (18 * 18 * 18);
    int a  = r2 / (18 * 18);
    int r3 = r2 % (18 * 18);
    int y  = r3 / 18, z = r3 % 18;
    int b = patch >> 4, slot = patch & 15;
    int p = idxarr[b * 16 + slot];
    int i = p >> 4, j = (p >> 2) & 3, k = p & 3;
    int X = i * 16 + a - 1, Y = j * 16 + y - 1, Z = k * 16 + z - 1;
    float v = 0.f;
    if ((unsigned)X < 64u && (unsigned)Y < 64u && (unsigned)Z < 64u)
        v = occ[(((size_t)(b * 2 + ch) * 64 + X) * 64 + Y) * 64 + Z];
    out[t] = v;
}

// ------------------------------------------------------------- local patches
// One thread per patch cell; writes all 16 channels (coalesced along z).
__global__ __launch_bounds__(256)
void local_patches_k(const int* __restrict__ grid, const float* __restrict__ feat,
                     const int* __restrict__ idxarr, float* __restrict__ out)
{
    int t = blockIdx.x * blockDim.x + threadIdx.x;
    if (t >= NPATCH * LOC_CELLS) return;
    int patch = t / LOC_CELLS;
    int cell  = t - patch * LOC_CELLS;
    int a  = cell / (36 * 36);
    int r  = cell % (36 * 36);
    int y  = r / 36, z = r % 36;
    int b = patch >> 4, slot = patch & 15;
    int p = idxarr[b * 16 + slot];
    int i = p >> 4, j = (p >> 2) & 3, k = p & 3;
    int X = i * 32 + a - 2, Y = j * 32 + y - 2, Z = k * 32 + z - 2;
    int src = -1;
    if ((unsigned)X < 128u && (unsigned)Y < 128u && (unsigned)Z < 128u)
        src = grid[(size_t)b * GRID_CELLS + ((X * GRID_R + Y) * GRID_R + Z)];
    size_t obase = (size_t)patch * (CFEAT * LOC_CELLS) + cell;
    if (src >= 0) {
        const float* f = feat + ((size_t)b * BPTS + src) * CFEAT;
        #pragma unroll
        for (int c = 0; c < CFEAT; ++c)
            out[obase + (size_t)c * LOC_CELLS] = f[c];
    } else {
        #pragma unroll
        for (int c = 0; c < CFEAT; ++c)
            out[obase + (size_t)c * LOC_CELLS] = 0.f;
    }
}

// --------------------------------------------------------------------- host
extern "C" void kernel_launch(void* const* d_in, const int* in_sizes, int n_in,
                              void* d_out, int out_size, void* d_ws, size_t ws_size,
                              hipStream_t stream)
{
    const float* curves   = (const float*)d_in[0];
    const float* surfaces = (const float*)d_in[1];
    const float* occ      = (const float*)d_in[2];
    const int*   idxarr   = (const int*)d_in[3];
    const float* W1       = (const float*)d_in[4];
    const float* b1       = (const float*)d_in[5];
    const float* W2       = (const float*)d_in[6];
    const float* b2       = (const float*)d_in[7];
    float* out = (float*)d_out;

    int*   grid = (int*)d_ws;                                   // 33.5 MB
    float* feat = (float*)((char*)d_ws + (size_t)NBATCH * GRID_CELLS * sizeof(int)); // 8 MB

    // 1) reset index grid to -1 (deterministic across graph replays)
    int n4 = NBATCH * GRID_CELLS / 4;
    init_grid_k<<<(n4 + 255) / 256, 256, 0, stream>>>((int4*)grid, n4);

    // 2) MLP (WMMA layer-2) + last-write-wins cell scatter
    int tiles  = NBATCH * BPTS / 16;       // 8192
    mlp_scatter_k<<<tiles / 4, 128, 0, stream>>>(curves, surfaces, W1, b1, W2, b2,
                                                 feat, grid);

    // 3) global patches (64 x 2 x 18^3)
    int ng = NPATCH * GLO_ELEMS;
    global_patches_k<<<(ng + 255) / 256, 256, 0, stream>>>(occ, idxarr, out);

    // 4) local patches (64 x 16 x 36^3)
    int nl = NPATCH * LOC_CELLS;
    local_patches_k<<<(nl + 255) / 256, 256, 0, stream>>>(grid, feat, idxarr,
                                                          out + (size_t)ng);
}